// CylinderGRUDecoder_90950227460810
// MI455X (gfx1250) — compile-verified
//
#include <hip/hip_runtime.h>
#include <math.h>

// ---------------- CDNA5 (gfx1250) wave32 WMMA types ----------------
typedef __attribute__((ext_vector_type(2))) float v2f;
typedef __attribute__((ext_vector_type(4))) float v4f;
typedef __attribute__((ext_vector_type(8))) float v8f;

#define WAVES_PER_BLK 8
#define TPB (WAVES_PER_BLK * 32)

#define HID 64
#define PFEAT 64
#define CIN 128              // HID + PFEAT
#define NITER 4
#define ROWSTRIDE 132        // CIN + 4 pad -> conflict-free A-frag ds_load_b64

#define D1 256
#define D2 256
#define D3 32
#define CH 32                // C_HALF
#define SPAT_I (D2 * D3)     // 8192
#define SPAT_J (D3)          // 32
#define CHSTRIDE ((size_t)D1 * D2 * D3)   // elements per channel slice
#define BSTRIDE (CHSTRIDE * CH)           // elements per batch per tensor

#define KB_STEPS (CIN / 4)                    // 32 K-steps of 4 for f32 WMMA
#define FRAG_F32_W64 (KB_STEPS * 4 * 32 * 2)  // 8192 floats (64-row weight)
#define FRAG_F32_W32 (KB_STEPS * 2 * 32 * 2)  // 4096 floats (Wd1, 32 rows)
#define FRAG_F32_ZR  (KB_STEPS * 4 * 32 * 4)  // 16384 floats (Wz|Wr interleaved)

// B-fragment layout for V_WMMA_F32_16X16X4_F32 (wave32):
//   lane L supplies B[k0][n], B[k0+1][n] with n = t*16 + L%16,
//   k0 = kb*4 + 2*(L/16)   (V0 = K0|K2 across lane halves, V1 = K1|K3).
__device__ inline void stage_frags(const float* __restrict__ W, float* frag,
                                   int ntiles, int tid) {
  const int total = KB_STEPS * ntiles * 32;   // float2 count
  for (int idx = tid; idx < total; idx += TPB) {
    const int lane = idx & 31;
    const int tile = (idx >> 5) % ntiles;
    const int kb   = idx / (32 * ntiles);
    const int col  = tile * 16 + (lane & 15);
    const int k0   = kb * 4 + ((lane >> 4) << 1);
    frag[idx * 2 + 0] = W[col * CIN + k0];
    frag[idx * 2 + 1] = W[col * CIN + k0 + 1];
  }
}

// Interleaved Wz|Wr fragments: one ds_load_b128 feeds both z- and r-WMMAs.
__device__ inline void stage_frags_zr(const float* __restrict__ Wz,
                                      const float* __restrict__ Wr,
                                      float* frag, int tid) {
  const int total = KB_STEPS * 4 * 32;   // float4 count
  for (int idx = tid; idx < total; idx += TPB) {
    const int lane = idx & 31;
    const int tile = (idx >> 5) & 3;
    const int kb   = idx >> 7;
    const int col  = tile * 16 + (lane & 15);
    const int k0   = kb * 4 + ((lane >> 4) << 1);
    frag[idx * 4 + 0] = Wz[col * CIN + k0];
    frag[idx * 4 + 1] = Wz[col * CIN + k0 + 1];
    frag[idx * 4 + 2] = Wr[col * CIN + k0];
    frag[idx * 4 + 3] = Wr[col * CIN + k0 + 1];
  }
}

// D(16 x NT*16) += A(16x128 from abuf rows) * B(frag).  K loop deliberately
// kept rolled (unroll 4) so B-fragment ds_loads stay inside the loop and are
// never hoisted/spilled to scratch across the GRU-iteration loop.
template <int NT>
__device__ inline void gemm_tiles(const float* abuf, const float* frag,
                                  int lane, v8f (&acc)[NT]) {
  const int row  = lane & 15;
  const int ksel = (lane >> 4) << 1;
#pragma unroll 4
  for (int kb = 0; kb < KB_STEPS; ++kb) {
    v2f a = *(const v2f*)(abuf + row * ROWSTRIDE + kb * 4 + ksel);
#pragma unroll
    for (int t = 0; t < NT; ++t) {
      v2f b = *(const v2f*)(frag + ((kb * NT + t) * 32 + lane) * 2);
      acc[t] = __builtin_amdgcn_wmma_f32_16x16x4_f32(
          false, a, false, b, (short)0, acc[t], false, false);
    }
  }
}

// Fused z & r GEMMs: shared A fragment, one b128 B-load feeding two WMMAs.
__device__ inline void gemm_zr(const float* abuf, const float* fzr, int lane,
                               v8f (&az)[4], v8f (&ar)[4]) {
  const int row  = lane & 15;
  const int ksel = (lane >> 4) << 1;
#pragma unroll 4
  for (int kb = 0; kb < KB_STEPS; ++kb) {
    v2f a = *(const v2f*)(abuf + row * ROWSTRIDE + kb * 4 + ksel);
#pragma unroll
    for (int t = 0; t < 4; ++t) {
      v4f zr = *(const v4f*)(fzr + ((kb * 4 + t) * 32 + lane) * 4);
      v2f bz = {zr.x, zr.y};
      v2f br = {zr.z, zr.w};
      az[t] = __builtin_amdgcn_wmma_f32_16x16x4_f32(
          false, a, false, bz, (short)0, az[t], false, false);
      ar[t] = __builtin_amdgcn_wmma_f32_16x16x4_f32(
          false, a, false, br, (short)0, ar[t], false, false);
    }
  }
}

__device__ inline float sigmoidf_(float x) { return 1.0f / (1.0f + expf(-x)); }

__global__ __launch_bounds__(TPB) void CylinderGRUDecoder_kernel(
    const float* __restrict__ before, const float* __restrict__ after,
    const float* __restrict__ px, const int* __restrict__ coords,
    const float* __restrict__ Wz, const float* __restrict__ bz,
    const float* __restrict__ Wr, const float* __restrict__ br,
    const float* __restrict__ Wq, const float* __restrict__ bq,
    const float* __restrict__ Wd1, const float* __restrict__ bd1,
    const float* __restrict__ Wd2, const float* __restrict__ bd2,
    float* __restrict__ out, int npts_total, int n_per_batch) {
  __shared__ float s_wzr[FRAG_F32_ZR];   // Wz|Wr interleaved (64 KB)
  __shared__ float s_wq[FRAG_F32_W64];   // 32 KB
  __shared__ float s_wd1[FRAG_F32_W32];  // 16 KB
  __shared__ float s_buf[WAVES_PER_BLK * 16 * ROWSTRIDE];  // 66 KB

  const int tid = threadIdx.x;
  stage_frags_zr(Wz, Wr, s_wzr, tid);
  stage_frags(Wq, s_wq, 4, tid);
  stage_frags(Wd1, s_wd1, 2, tid);
  __syncthreads();   // only cross-wave sync; all later traffic is wave-private

  const int lane = tid & 31;
  const int wv   = tid >> 5;
  const int tile = blockIdx.x * WAVES_PER_BLK + wv;
  const int ntiles = (npts_total + 15) >> 4;
  if (tile >= ntiles) return;   // wave-uniform; EXEC stays all-1s for WMMA

  float* buf = s_buf + wv * (16 * ROWSTRIDE);
  const int p0 = tile * 16;

  // ---- gather h = [before_pt | after_pt] and x into row-major staging ----
  for (int p = 0; p < 16; ++p) {
    int pg = p0 + p;
    if (pg >= npts_total) pg = npts_total - 1;  // safe pad rows
    const int b = pg / n_per_batch;
    const int ci = coords[pg * 3 + 0];
    const int cj = coords[pg * 3 + 1];
    const int ck = coords[pg * 3 + 2];
    const size_t spat = (size_t)ci * SPAT_I + (size_t)cj * SPAT_J + (size_t)ck;
    const size_t gidx = (size_t)b * BSTRIDE + (size_t)lane * CHSTRIDE + spat;
    buf[p * ROWSTRIDE + lane]       = before[gidx];
    buf[p * ROWSTRIDE + 32 + lane]  = after[gidx];
    buf[p * ROWSTRIDE + 64 + lane]  = px[(size_t)pg * PFEAT + lane];
    buf[p * ROWSTRIDE + 96 + lane]  = px[(size_t)pg * PFEAT + 32 + lane];
  }

  // per-lane biases in C/D layout (N = t*16 + lane%16)
  const int colL = lane & 15;
  float bzt[4], brt[4], bqt[4];
#pragma unroll
  for (int t = 0; t < 4; ++t) {
    bzt[t] = bz[t * 16 + colL];
    brt[t] = br[t * 16 + colL];
    bqt[t] = bq[t * 16 + colL];
  }
  const float bd1t[2] = {bd1[colL], bd1[16 + colL]};
  const int mrow = (lane >> 4) << 3;  // C/D layout: element (v,lane) is M=v+mrow

  // ---------------- GRU iterations ----------------
#pragma unroll 1
  for (int it = 0; it < NITER; ++it) {
    v8f accz[4], accr[4];
#pragma unroll
    for (int t = 0; t < 4; ++t) {
      accz[t] = (v8f){0, 0, 0, 0, 0, 0, 0, 0};
      accr[t] = (v8f){0, 0, 0, 0, 0, 0, 0, 0};
    }
    gemm_zr(buf, s_wzr, lane, accz, accr);

    float hreg[4][8];
#pragma unroll
    for (int t = 0; t < 4; ++t) {
#pragma unroll
      for (int v = 0; v < 8; ++v) {
        accz[t][v] = sigmoidf_(accz[t][v] + bzt[t]);       // z (kept)
        const float rv = sigmoidf_(accr[t][v] + brt[t]);   // r
        const float hv = buf[(v + mrow) * ROWSTRIDE + t * 16 + colL];
        hreg[t][v] = hv;
        accr[t][v] = rv * hv;                              // r*h
      }
    }
    // overwrite h-region with r*h (rhx = [r*h | x]); same-wave DS ordering
#pragma unroll
    for (int t = 0; t < 4; ++t)
#pragma unroll
      for (int v = 0; v < 8; ++v)
        buf[(v + mrow) * ROWSTRIDE + t * 16 + colL] = accr[t][v];

    v8f accq[4];
#pragma unroll
    for (int t = 0; t < 4; ++t) accq[t] = (v8f){0, 0, 0, 0, 0, 0, 0, 0};
    gemm_tiles<4>(buf, s_wq, lane, accq);

#pragma unroll
    for (int t = 0; t < 4; ++t)
#pragma unroll
      for (int v = 0; v < 8; ++v) {
        const float qv = tanhf(accq[t][v] + bqt[t]);
        const float zv = accz[t][v];
        buf[(v + mrow) * ROWSTRIDE + t * 16 + colL] =
            (1.0f - zv) * hreg[t][v] + zv * qv;            // new h
      }
  }

  // ---------------- decoder ----------------
  v8f accd[2];
#pragma unroll
  for (int t = 0; t < 2; ++t) accd[t] = (v8f){0, 0, 0, 0, 0, 0, 0, 0};
  gemm_tiles<2>(buf, s_wd1, lane, accd);

#pragma unroll
  for (int t = 0; t < 2; ++t)
#pragma unroll
    for (int v = 0; v < 8; ++v) {
      const float xv = accd[t][v] + bd1t[t];
      const float g = 0.5f * xv * (1.0f + erff(xv * 0.70710678118654752f));
      buf[(v + mrow) * ROWSTRIDE + t * 16 + colL] = g;   // hmid -> cols 0..31
    }

  // tiny 32 -> 3 contraction, one point per lane (lanes 0..15)
  if (lane < 16) {
    const int p = lane;
    if (p0 + p < npts_total) {
      const float* hm = buf + p * ROWSTRIDE;
      float f0 = bd2[0], f1 = bd2[1], f2 = bd2[2];
#pragma unroll
      for (int k = 0; k < 32; ++k) {
        const float hv = hm[k];
        f0 += hv * Wd2[k];
        f1 += hv * Wd2[32 + k];
        f2 += hv * Wd2[64 + k];
      }
      const size_t o = (size_t)(p0 + p) * 3;
      out[o + 0] = f0;
      out[o + 1] = f1;
      out[o + 2] = f2;
    }
  }
}

extern "C" void kernel_launch(void* const* d_in, const int* in_sizes, int n_in,
                              void* d_out, int out_size, void* d_ws,
                              size_t ws_size, hipStream_t stream) {
  const float* before = (const float*)d_in[0];
  const float* after  = (const float*)d_in[1];
  const float* px     = (const float*)d_in[2];
  const int*   coords = (const int*)d_in[3];
  const float* Wz  = (const float*)d_in[4];
  const float* bz  = (const float*)d_in[5];
  const float* Wr  = (const float*)d_in[6];
  const float* br  = (const float*)d_in[7];
  const float* Wq  = (const float*)d_in[8];
  const float* bq  = (const float*)d_in[9];
  const float* Wd1 = (const float*)d_in[10];
  const float* bd1 = (const float*)d_in[11];
  const float* Wd2 = (const float*)d_in[12];
  const float* bd2 = (const float*)d_in[13];
  float* out = (float*)d_out;

  const int npts_total  = in_sizes[2] / PFEAT;                 // B*N
  const int B           = in_sizes[0] / (int)(CH * CHSTRIDE);  // batches
  const int n_per_batch = npts_total / (B > 0 ? B : 1);

  const int ntiles = (npts_total + 15) / 16;
  const int blocks = (ntiles + WAVES_PER_BLK - 1) / WAVES_PER_BLK;

  CylinderGRUDecoder_kernel<<<blocks, TPB, 0, stream>>>(
      before, after, px, coords, Wz, bz, Wr, br, Wq, bq, Wd1, bd1, Wd2, bd2,
      out, npts_total, n_per_batch);
}